// NodeAttentionHead_37426345017542
// MI455X (gfx1250) — compile-verified
//
#include <hip/hip_runtime.h>

// Problem constants (from the reference)
#define N_NODES 50000
#define DEG     16
#define NIN     128
#define NOUT    64

typedef __attribute__((ext_vector_type(2))) float v2f;
typedef __attribute__((ext_vector_type(8))) float v8f;

// -------------------------------------------------------------------------
// Kernel 1: h_v = node_fts @ W_node, e_v = edge_fts @ W_edge  (f32 WMMA)
// One wave computes a 16x64 tile: 4 accumulators, K-loop of 32 steps of
// V_WMMA_F32_16X16X4_F32. A-fragment (16x4 f32) is reused across 4 N-tiles.
// A layout: lane m(0..15), half kh: a.x=A[m][k0+2kh], a.y=A[m][k0+2kh+1]
// B layout mirrors A with N=lane; C/D: vgpr r -> row r + 8*kh, col = lane&15
// -------------------------------------------------------------------------
__global__ __launch_bounds__(256) void proj_wmma_kernel(
    const float* __restrict__ node_fts, const float* __restrict__ edge_fts,
    const float* __restrict__ W_node,   const float* __restrict__ W_edge,
    float* __restrict__ hv, float* __restrict__ ev)
{
    const int wave   = (blockIdx.x * blockDim.x + threadIdx.x) >> 5;
    const int ntiles = N_NODES / 16;            // 3125 (N divisible by 16)
    if (wave >= 2 * ntiles) return;             // wave-uniform exit (EXEC stays full)
    const int  mat  = (wave >= ntiles) ? 1 : 0;
    const int  tile = mat ? (wave - ntiles) : wave;
    const float* A  = mat ? edge_fts : node_fts;
    const float* W  = mat ? W_edge   : W_node;
    float*       O  = mat ? ev       : hv;

    const int row0 = tile * 16;
    const int lane = threadIdx.x & 31;
    const int m    = lane & 15;
    const int kh   = lane >> 4;                 // 0 or 1

    const float* arow = A + (size_t)(row0 + m) * NIN + 2 * kh;
    v8f acc0 = {}, acc1 = {}, acc2 = {}, acc3 = {};

    #pragma unroll 4
    for (int k0 = 0; k0 < NIN; k0 += 4) {
        v2f a = *(const v2f*)(arow + k0);
        const float* w0 = W + (size_t)(k0 + 2 * kh) * NOUT + m;
        const float* w1 = w0 + NOUT;
        v2f b0 = { w0[0],  w1[0]  };
        v2f b1 = { w0[16], w1[16] };
        v2f b2 = { w0[32], w1[32] };
        v2f b3 = { w0[48], w1[48] };
        acc0 = __builtin_amdgcn_wmma_f32_16x16x4_f32(false, a, false, b0, (short)0, acc0, false, false);
        acc1 = __builtin_amdgcn_wmma_f32_16x16x4_f32(false, a, false, b1, (short)0, acc1, false, false);
        acc2 = __builtin_amdgcn_wmma_f32_16x16x4_f32(false, a, false, b2, (short)0, acc2, false, false);
        acc3 = __builtin_amdgcn_wmma_f32_16x16x4_f32(false, a, false, b3, (short)0, acc3, false, false);
    }

    float* orow = O + (size_t)(row0 + kh * 8) * NOUT + m;
    #pragma unroll
    for (int r = 0; r < 8; ++r) {
        orow[(size_t)r * NOUT +  0] = acc0[r];
        orow[(size_t)r * NOUT + 16] = acc1[r];
        orow[(size_t)r * NOUT + 32] = acc2[r];
        orow[(size_t)r * NOUT + 48] = acc3[r];
    }
}

// -------------------------------------------------------------------------
// Kernel 2: per-node attention scalars.
// alpha_src[i] = h_v[i]·a_node[0:64], alpha_dst[i] = h_v[i]·a_node[64:128]
// beta_* likewise with e_v / a_edge. One wave per node, shfl reduction.
// -------------------------------------------------------------------------
__global__ __launch_bounds__(256) void alpha_kernel(
    const float* __restrict__ hv, const float* __restrict__ ev,
    const float* __restrict__ a_node, const float* __restrict__ a_edge,
    float* __restrict__ asrc, float* __restrict__ adst,
    float* __restrict__ bsrc, float* __restrict__ bdst)
{
    const int wave = (blockIdx.x * blockDim.x + threadIdx.x) >> 5;
    if (wave >= N_NODES) return;
    const int lane = threadIdx.x & 31;

    const float a0l = a_node[lane],      a0h = a_node[lane + 32];
    const float a1l = a_node[lane + 64], a1h = a_node[lane + 96];
    const float b0l = a_edge[lane],      b0h = a_edge[lane + 32];
    const float b1l = a_edge[lane + 64], b1h = a_edge[lane + 96];

    const float hlo = hv[(size_t)wave * NOUT + lane];
    const float hhi = hv[(size_t)wave * NOUT + 32 + lane];
    const float elo = ev[(size_t)wave * NOUT + lane];
    const float ehi = ev[(size_t)wave * NOUT + 32 + lane];

    float ps = hlo * a0l + hhi * a0h;
    float pd = hlo * a1l + hhi * a1h;
    float qs = elo * b0l + ehi * b0h;
    float qd = elo * b1l + ehi * b1h;
    #pragma unroll
    for (int off = 16; off; off >>= 1) {
        ps += __shfl_xor(ps, off, 32);
        pd += __shfl_xor(pd, off, 32);
        qs += __shfl_xor(qs, off, 32);
        qd += __shfl_xor(qd, off, 32);
    }
    if (lane == 0) {
        asrc[wave] = ps; adst[wave] = pd;
        bsrc[wave] = qs; bdst[wave] = qd;
    }
}

// -------------------------------------------------------------------------
// Kernel 3: per-node softmax over its 16 edges + gather-accumulate.
// One wave per node; lanes 0..15 own the 16 edges of segment i; all 32
// lanes own 2 output columns each (float2). Block writes 4 partial sums
// (sum / sumsq of normalized attentions) for the deterministic variance.
// -------------------------------------------------------------------------
__global__ __launch_bounds__(256) void edge_kernel(
    const float* __restrict__ hv, const float* __restrict__ ev,
    const int*  __restrict__ edges,
    const float* __restrict__ asrc, const float* __restrict__ adst,
    const float* __restrict__ bsrc, const float* __restrict__ bdst,
    float* __restrict__ node_out, float* __restrict__ edge_out,
    float* __restrict__ partials)
{
    __shared__ float sred[4][8];
    const int lane = threadIdx.x & 31;
    const int wid  = threadIdx.x >> 5;
    const int node = blockIdx.x * 8 + wid;

    float att_n = 0.f, att_e = 0.f;
    int   d = 0;
    if (node < N_NODES && lane < 16) {
        int2 ed = ((const int2*)edges)[(size_t)node * DEG + lane];
        const int s = ed.x;
        d = ed.y;
        float xn = asrc[s] + adst[d];
        float xe = bsrc[s] + bdst[d];
        xn = xn > 0.f ? xn : 0.2f * xn;            // leaky relu
        xe = xe > 0.f ? xe : 0.2f * xe;
        xn = fminf(fmaxf(xn, -2.f), 2.f);          // clip
        xe = fminf(fmaxf(xe, -2.f), 2.f);
        att_n = __expf(xn);
        att_e = __expf(xe);
    }

    float sum_n = att_n, sum_e = att_e;
    #pragma unroll
    for (int off = 16; off; off >>= 1) {
        sum_n += __shfl_xor(sum_n, off, 32);
        sum_e += __shfl_xor(sum_e, off, 32);
    }
    const bool act = (node < N_NODES) && (lane < 16);
    float norm_n = act ? att_n / sum_n : 0.f;
    float norm_e = act ? att_e / sum_e : 0.f;

    // wave partials for variance (sum and sum of squares)
    float pn = norm_n, pn2 = norm_n * norm_n, pe = norm_e, pe2 = norm_e * norm_e;
    #pragma unroll
    for (int off = 16; off; off >>= 1) {
        pn  += __shfl_xor(pn,  off, 32);
        pn2 += __shfl_xor(pn2, off, 32);
        pe  += __shfl_xor(pe,  off, 32);
        pe2 += __shfl_xor(pe2, off, 32);
    }
    if (lane == 0) { sred[0][wid] = pn; sred[1][wid] = pn2; sred[2][wid] = pe; sred[3][wid] = pe2; }
    __syncthreads();
    if (threadIdx.x < 4) {
        float t = 0.f;
        #pragma unroll
        for (int w = 0; w < 8; ++w) t += sred[threadIdx.x][w];
        partials[(size_t)blockIdx.x * 4 + threadIdx.x] = t;
    }
    if (node >= N_NODES) return;

    // gather-accumulate dst halves (h_v/e_v are L2-resident: 25.6 MB total)
    const float2* hv2 = (const float2*)hv;
    const float2* ev2 = (const float2*)ev;
    float2 acc_h = {0.f, 0.f}, acc_e = {0.f, 0.f};
    #pragma unroll
    for (int j = 0; j < 16; ++j) {
        const int   dj = __shfl(d,      j, 32);
        const float wn = __shfl(norm_n, j, 32);
        const float we = __shfl(norm_e, j, 32);
        const float2 hvv = hv2[(size_t)dj * 32 + lane];
        const float2 evv = ev2[(size_t)dj * 32 + lane];
        acc_h.x += wn * hvv.x;  acc_h.y += wn * hvv.y;
        acc_e.x += we * evv.x;  acc_e.y += we * evv.y;
    }

    const float2 hs = hv2[(size_t)node * 32 + lane];
    const float2 es = ev2[(size_t)node * 32 + lane];
    float2* no2 = (float2*)node_out;
    float2* eo2 = (float2*)edge_out;
    // src half: sum_j norm_j * h_v[src_j=node] = (sum norm) * h_v[node]
    float2 o0 = { pn * hs.x, pn * hs.y };
    float2 p0 = { pe * es.x, pe * es.y };
    no2[(size_t)node * 64 + lane]      = o0;
    no2[(size_t)node * 64 + 32 + lane] = acc_h;
    eo2[(size_t)node * 64 + lane]      = p0;
    eo2[(size_t)node * 64 + 32 + lane] = acc_e;
}

// -------------------------------------------------------------------------
// Kernel 4: deterministic final reduction of block partials (double) and
// variance with ddof=1: var = (sum_sq - sum^2/E) / (E-1).
// -------------------------------------------------------------------------
__global__ __launch_bounds__(256) void finalize_kernel(
    const float* __restrict__ partials, int nblocks, float* __restrict__ outv)
{
    __shared__ double red[4][256];
    double s0 = 0.0, s1 = 0.0, s2 = 0.0, s3 = 0.0;
    for (int i = threadIdx.x; i < nblocks; i += 256) {
        s0 += (double)partials[(size_t)i * 4 + 0];
        s1 += (double)partials[(size_t)i * 4 + 1];
        s2 += (double)partials[(size_t)i * 4 + 2];
        s3 += (double)partials[(size_t)i * 4 + 3];
    }
    red[0][threadIdx.x] = s0; red[1][threadIdx.x] = s1;
    red[2][threadIdx.x] = s2; red[3][threadIdx.x] = s3;
    __syncthreads();
    for (int off = 128; off; off >>= 1) {
        if (threadIdx.x < off) {
            red[0][threadIdx.x] += red[0][threadIdx.x + off];
            red[1][threadIdx.x] += red[1][threadIdx.x + off];
            red[2][threadIdx.x] += red[2][threadIdx.x + off];
            red[3][threadIdx.x] += red[3][threadIdx.x + off];
        }
        __syncthreads();
    }
    if (threadIdx.x == 0) {
        const double E = (double)N_NODES * (double)DEG;
        const double var_n = (red[1][0] - red[0][0] * red[0][0] / E) / (E - 1.0);
        const double var_e = (red[3][0] - red[2][0] * red[2][0] / E) / (E - 1.0);
        outv[0] = (float)var_n;
        outv[1] = (float)var_e;
    }
}

// -------------------------------------------------------------------------
// Launch: inputs (node_fts, edge_fts, edges, W_node, W_edge, a_node, a_edge)
// d_out = [node_out (N*128) | edge_out (N*128) | var_n | var_e]
// d_ws  = [h_v (N*64) | e_v (N*64) | asrc | adst | bsrc | bdst | partials]
//         = ~26.5 MB of scratch
// -------------------------------------------------------------------------
extern "C" void kernel_launch(void* const* d_in, const int* in_sizes, int n_in,
                              void* d_out, int out_size, void* d_ws, size_t ws_size,
                              hipStream_t stream) {
    const float* node_fts = (const float*)d_in[0];
    const float* edge_fts = (const float*)d_in[1];
    const int*   edges    = (const int*)  d_in[2];
    const float* W_node   = (const float*)d_in[3];
    const float* W_edge   = (const float*)d_in[4];
    const float* a_node   = (const float*)d_in[5];
    const float* a_edge   = (const float*)d_in[6];

    float* out      = (float*)d_out;
    float* node_out = out;
    float* edge_out = out + (size_t)N_NODES * 128;
    float* outv     = out + (size_t)2 * N_NODES * 128;

    float* ws   = (float*)d_ws;
    float* hv   = ws;
    float* ev   = hv + (size_t)N_NODES * NOUT;
    float* asrc = ev + (size_t)N_NODES * NOUT;
    float* adst = asrc + N_NODES;
    float* bsrc = adst + N_NODES;
    float* bdst = bsrc + N_NODES;
    float* partials = bdst + N_NODES;

    const int ntiles = N_NODES / 16;                 // 3125
    const int wavesA = 2 * ntiles;                   // 6250
    const int blocksA = (wavesA + 7) / 8;            // 8 waves per block
    proj_wmma_kernel<<<blocksA, 256, 0, stream>>>(node_fts, edge_fts, W_node, W_edge, hv, ev);

    const int blocksB = (N_NODES + 7) / 8;           // one wave per node
    alpha_kernel<<<blocksB, 256, 0, stream>>>(hv, ev, a_node, a_edge, asrc, adst, bsrc, bdst);

    const int blocksC = (N_NODES + 7) / 8;           // 6250, one wave per node
    edge_kernel<<<blocksC, 256, 0, stream>>>(hv, ev, edges, asrc, adst, bsrc, bdst,
                                             node_out, edge_out, partials);

    finalize_kernel<<<1, 256, 0, stream>>>(partials, blocksC, outv);
}